// GNNLayer_82678120448824
// MI455X (gfx1250) — compile-verified
//
#include <hip/hip_runtime.h>
#include <hip/hip_bf16.h>

// Problem constants (from reference)
#define E_CNT 500000
#define N_CNT 200000
#define D_DIM 128
#define A_DIM 64
#define B_CNT 32

// Output layout (flat, reference tuple order)
static constexpr long long OFF_NN    = 0;                       // num_node (2)
static constexpr long long OFF_NE    = 2;                       // num_edge (2)
static constexpr long long OFF_H     = 4;                       // hidden_new N*128
static constexpr long long OFF_ALPHA = OFF_H + (long long)N_CNT * D_DIM;
static constexpr long long OFF_NODES = OFF_ALPHA + E_CNT;       // nodes N*2
static constexpr long long OFF_EDGES = OFF_NODES + (long long)N_CNT * 2;
static constexpr long long OFF_OLD   = OFF_EDGES + (long long)E_CNT * 6;

// Packed-weight element counts (float4 units)
#define EPACK_N (4 * 32 * A_DIM)    // 8192 float4 = 128 KB : [s*32+k4][n<64]
#define NPACK_N (32 * D_DIM)        // 4096 float4 =  64 KB : [k4][n<128]

typedef float v2f __attribute__((ext_vector_type(2)));
typedef float v8f __attribute__((ext_vector_type(8)));

__device__ __forceinline__ void atomic_add_f32(float* p, float v) {
    __hip_atomic_fetch_add(p, v, __ATOMIC_RELAXED, __HIP_MEMORY_SCOPE_AGENT);
}

// ---------------------------------------------------------------------------
// Pass-through outputs: scalars, nodes (int32), edges (int64), old idx (int32)
// ---------------------------------------------------------------------------
__global__ void passthrough_kernel(const int* __restrict__ nodes,
                                   const long long* __restrict__ edges,
                                   const int* __restrict__ oldn,
                                   float* __restrict__ out) {
    const long long N2 = (long long)N_CNT * 2;
    const long long E6 = (long long)E_CNT * 6;
    const long long total = 4 + N2 + E6 + B_CNT;
    long long i = (long long)blockIdx.x * blockDim.x + threadIdx.x;
    const long long stride = (long long)gridDim.x * blockDim.x;
    for (; i < total; i += stride) {
        if (i < 2) {
            out[OFF_NN + i] = (float)N_CNT / (float)B_CNT;
        } else if (i < 4) {
            out[OFF_NE + (i - 2)] = (float)E_CNT / (float)B_CNT;
        } else if (i < 4 + N2) {
            long long j = i - 4;
            out[OFF_NODES + j] = (float)nodes[j];
        } else if (i < 4 + N2 + E6) {
            long long j = i - 4 - N2;
            out[OFF_EDGES + j] = (float)edges[j];
        } else {
            long long j = i - 4 - N2 - E6;
            out[OFF_OLD + j] = (float)oldn[j];
        }
    }
}

// ---------------------------------------------------------------------------
// Weight repack: float4 {W[4k4+0][n], W[4k4+1][n], W[4k4+2][n], W[4k4+3][n]}
// so a WMMA B-fragment is one 8-byte read at float-offset (idx*4 + 2*half).
// ---------------------------------------------------------------------------
__global__ void pack_edge_weights(const float* __restrict__ Ws,
                                  const float* __restrict__ Wr,
                                  const float* __restrict__ Wq,
                                  const float* __restrict__ Wqr,
                                  float4* __restrict__ out) {
    const int i = blockIdx.x * blockDim.x + threadIdx.x;
    if (i >= EPACK_N) return;
    const int n   = i & (A_DIM - 1);
    const int kk4 = i >> 6;            // s*32 + k4
    const int s   = kk4 >> 5;
    const int k   = (kk4 & 31) * 4;
    const float* W = (s == 0) ? Ws : (s == 1) ? Wr : (s == 2) ? Wq : Wqr;
    out[i] = make_float4(W[(k + 0) * A_DIM + n], W[(k + 1) * A_DIM + n],
                         W[(k + 2) * A_DIM + n], W[(k + 3) * A_DIM + n]);
}

__global__ void pack_node_weights(const float* __restrict__ Wh,
                                  float4* __restrict__ out) {
    const int i = blockIdx.x * blockDim.x + threadIdx.x;
    if (i >= NPACK_N) return;
    const int n  = i & (D_DIM - 1);
    const int k  = (i >> 7) * 4;
    out[i] = make_float4(Wh[(k + 0) * D_DIM + n], Wh[(k + 1) * D_DIM + n],
                         Wh[(k + 2) * D_DIM + n], Wh[(k + 3) * D_DIM + n]);
}

// ---------------------------------------------------------------------------
// Edge kernel: per 16-edge tile (one wave32):
//   pre[16x64] = [hs|hr|hqr|hr*hqr] (16x512) @ [Ws;Wr;Wq;Wqr] (512x64) + Ws_b
//   alpha = sigmoid(relu(pre) @ walpha + b)         -> d_out alpha section
//   agg[obj] += alpha * hs * hr                     -> fp32 global atomics
// Weights staged once per block in LDS (128 KB); B-frags are ds_load_b64,
// bank-conflict-free (lanes cover float-banks 4r+{0,1} / 4r+{2,3}).
// ---------------------------------------------------------------------------
__global__ __launch_bounds__(256) void edge_kernel(
    const float* __restrict__ q_emb, const float* __restrict__ rela,
    const float* __restrict__ hidden, const long long* __restrict__ edges,
    const float4* __restrict__ packW, const float* __restrict__ Wsb,
    const float* __restrict__ walW, const float* __restrict__ walb,
    float* __restrict__ outAlpha, float* __restrict__ agg) {
    __shared__ float4 sw[EPACK_N];                 // 128 KB packed weights
    for (int i = threadIdx.x; i < EPACK_N; i += 256) sw[i] = packW[i];
    __syncthreads();
    const float* lw = (const float*)sw;

    const int lane = threadIdx.x & 31;
    const int wave = threadIdx.x >> 5;
    const int half = lane >> 4;     // K-half for A/B fragments
    const int row  = lane & 15;     // M row (A) / N column (B,C)
    const long long tile  = (long long)blockIdx.x * 8 + wave;
    const long long eBase = tile * 16;
    if (eBase >= E_CNT) return;

    // lanes 0..15 load the 4 int64 indices of their row; others idle
    int sub_l = 0, rel_l = 0, b_l = 0, obj_l = 0;
    if (lane < 16) {
        long long e = eBase + row;
        if (e >= E_CNT) e = E_CNT - 1;
        const long long* er = edges + e * 6;
        b_l   = (int)er[0];
        rel_l = (int)er[2];
        sub_l = (int)er[4];
        obj_l = (int)er[5];
    }
    // every lane needs the indices of row (lane&15)
    const int sub_me = __shfl(sub_l, row, 32);
    const int rel_me = __shfl(rel_l, row, 32);
    const int b_me   = __shfl(b_l,   row, 32);

    const float* hs_row = hidden + (size_t)sub_me * D_DIM;
    const float* hr_row = rela   + (size_t)rel_me * D_DIM;
    const float* hq_row = q_emb  + (size_t)b_me   * D_DIM;

    v8f c0 = {}, c1 = {}, c2 = {}, c3 = {};   // 4 N-tiles of pre (16x64)
    const int k0h = 2 * half;

    #pragma unroll
    for (int s = 0; s < 4; ++s) {
        const float* fa = (s == 0) ? hs_row : (s == 2) ? hq_row : hr_row;
        for (int k4 = 0; k4 < 32; ++k4) {
            const int k = k4 * 4 + k0h;
            v2f a;
            if (s < 3) {
                a = *(const v2f*)(fa + k);
            } else {                       // hr * h_qr on the fly
                v2f x = *(const v2f*)(hr_row + k);
                v2f y = *(const v2f*)(hq_row + k);
                a.x = x.x * y.x; a.y = x.y * y.y;
            }
            // packed B-frag: one b64 LDS read per WMMA
            const float* wp = lw + ((((s * 32 + k4) * A_DIM + row) << 2) + k0h);
            v2f b0 = *(const v2f*)(wp);
            v2f b1 = *(const v2f*)(wp + 16 * 4);
            v2f b2 = *(const v2f*)(wp + 32 * 4);
            v2f b3 = *(const v2f*)(wp + 48 * 4);
            c0 = __builtin_amdgcn_wmma_f32_16x16x4_f32(false, a, false, b0, (short)0, c0, false, false);
            c1 = __builtin_amdgcn_wmma_f32_16x16x4_f32(false, a, false, b1, (short)0, c1, false, false);
            c2 = __builtin_amdgcn_wmma_f32_16x16x4_f32(false, a, false, b2, (short)0, c2, false, false);
            c3 = __builtin_amdgcn_wmma_f32_16x16x4_f32(false, a, false, b3, (short)0, c3, false, false);
        }
    }

    // bias (per output column) + relu + attention dot with walpha
    const float bias0 = Wsb[row],      bias1 = Wsb[16 + row];
    const float bias2 = Wsb[32 + row], bias3 = Wsb[48 + row];
    const float wa0 = walW[row],      wa1 = walW[16 + row];
    const float wa2 = walW[32 + row], wa3 = walW[48 + row];
    const float ab = walb[0];

    float alphaR[8];  // lanes<16: rows 0..7; lanes>=16: rows 8..15
    #pragma unroll
    for (int i = 0; i < 8; ++i) {
        float p = fmaxf(c0[i] + bias0, 0.f) * wa0
                + fmaxf(c1[i] + bias1, 0.f) * wa1
                + fmaxf(c2[i] + bias2, 0.f) * wa2
                + fmaxf(c3[i] + bias3, 0.f) * wa3;
        p += __shfl_xor(p, 1, 32);
        p += __shfl_xor(p, 2, 32);
        p += __shfl_xor(p, 4, 32);
        p += __shfl_xor(p, 8, 32);   // full 16-lane half reduction
        alphaR[i] = 1.0f / (1.0f + __expf(-(p + ab)));
    }

    // message scatter: agg[obj] += alpha * hs * hr (32 lanes x float4 = 128)
    #pragma unroll
    for (int r = 0; r < 16; ++r) {
        const long long e = eBase + r;
        if (e < E_CNT) {
            const float al = __shfl(alphaR[r & 7], (r < 8) ? 0 : 16, 32);
            const int sub_r = __shfl(sub_l, r, 32);
            const int rel_r = __shfl(rel_l, r, 32);
            const int obj_r = __shfl(obj_l, r, 32);
            if (lane == 0) outAlpha[e] = al;
            const float4 hs4 = *(const float4*)(hidden + (size_t)sub_r * D_DIM + lane * 4);
            const float4 hr4 = *(const float4*)(rela   + (size_t)rel_r * D_DIM + lane * 4);
            float* dst = agg + (size_t)obj_r * D_DIM + lane * 4;
            atomic_add_f32(dst + 0, al * hs4.x * hr4.x);
            atomic_add_f32(dst + 1, al * hs4.y * hr4.y);
            atomic_add_f32(dst + 2, al * hs4.z * hr4.z);
            atomic_add_f32(dst + 3, al * hs4.w * hr4.w);
        }
    }
}

// ---------------------------------------------------------------------------
// Node kernel: hidden_new = agg (N x 128) @ Wh_W (128 x 128), fp32 WMMA.
// One wave per 16-node tile; Wh staged packed in LDS (64 KB per block).
// ---------------------------------------------------------------------------
__global__ __launch_bounds__(256) void node_kernel(
    const float* __restrict__ agg, const float4* __restrict__ packW,
    float* __restrict__ outHidden) {
    __shared__ float4 sw[NPACK_N];                 // 64 KB packed Wh
    for (int i = threadIdx.x; i < NPACK_N; i += 256) sw[i] = packW[i];
    __syncthreads();
    const float* lw = (const float*)sw;

    const int lane = threadIdx.x & 31;
    const int wave = threadIdx.x >> 5;
    const int half = lane >> 4;
    const int row  = lane & 15;
    const long long tile = (long long)blockIdx.x * 8 + wave;
    const long long m0 = tile * 16;
    if (m0 >= N_CNT) return;

    const float* arow = agg + (size_t)(m0 + row) * D_DIM;
    v8f c[8] = {};
    const int k0h = 2 * half;
    for (int k4 = 0; k4 < 32; ++k4) {
        const v2f a = *(const v2f*)(arow + k4 * 4 + k0h);
        const float* wp = lw + (((k4 * D_DIM + row) << 2) + k0h);
        #pragma unroll
        for (int t = 0; t < 8; ++t) {
            v2f b = *(const v2f*)(wp + t * 16 * 4);
            c[t] = __builtin_amdgcn_wmma_f32_16x16x4_f32(false, a, false, b, (short)0, c[t], false, false);
        }
    }
    #pragma unroll
    for (int t = 0; t < 8; ++t) {
        #pragma unroll
        for (int i = 0; i < 8; ++i) {
            const long long m = m0 + i + 8 * half;
            outHidden[(size_t)m * D_DIM + t * 16 + row] = c[t][i];
        }
    }
}

// ---------------------------------------------------------------------------
extern "C" void kernel_launch(void* const* d_in, const int* in_sizes, int n_in,
                              void* d_out, int out_size, void* d_ws, size_t ws_size,
                              hipStream_t stream) {
    (void)in_sizes; (void)n_in; (void)out_size; (void)ws_size;
    const float*     q_emb  = (const float*)d_in[2];
    const float*     rela   = (const float*)d_in[3];
    const float*     hidden = (const float*)d_in[4];
    const long long* edges  = (const long long*)d_in[5];
    const int*       nodes  = (const int*)d_in[6];
    const int*       oldn   = (const int*)d_in[7];
    const float*     WsW    = (const float*)d_in[8];
    const float*     Wsb    = (const float*)d_in[9];
    const float*     WrW    = (const float*)d_in[10];
    const float*     WqW    = (const float*)d_in[11];
    const float*     WqrW   = (const float*)d_in[12];
    const float*     walW   = (const float*)d_in[13];
    const float*     walb   = (const float*)d_in[14];
    const float*     WhW    = (const float*)d_in[15];

    float* out = (float*)d_out;
    // workspace layout: [agg N*128 f32][packed edge W][packed node W]
    float*  agg   = (float*)d_ws;
    const size_t aggBytes = (size_t)N_CNT * D_DIM * sizeof(float);
    float4* packE = (float4*)((char*)d_ws + aggBytes);
    float4* packN = packE + EPACK_N;

    // zero the segment-sum accumulator (capturable async memset node)
    hipMemsetAsync(agg, 0, aggBytes, stream);

    // one-time weight repack into WMMA-fragment-friendly float4 layout
    pack_edge_weights<<<(EPACK_N + 255) / 256, 256, 0, stream>>>(WsW, WrW, WqW, WqrW, packE);
    pack_node_weights<<<(NPACK_N + 255) / 256, 256, 0, stream>>>(WhW, packN);

    // pass-through sections (independent of compute)
    passthrough_kernel<<<2048, 256, 0, stream>>>(nodes, edges, oldn, out);

    // edge stage: pre/alpha/message-scatter
    const int edgeWaves  = (E_CNT + 15) / 16;       // 31250 tiles
    const int edgeBlocks = (edgeWaves + 7) / 8;     // 8 waves / 256-thread block
    edge_kernel<<<edgeBlocks, 256, 0, stream>>>(q_emb, rela, hidden, edges,
                                                packE, Wsb, walW, walb,
                                                out + OFF_ALPHA, agg);

    // node stage: hidden_new = agg @ Wh
    const int nodeWaves  = (N_CNT + 15) / 16;       // 12500 tiles
    const int nodeBlocks = (nodeWaves + 7) / 8;
    node_kernel<<<nodeBlocks, 256, 0, stream>>>(agg, packN, out + OFF_H);
}